// GNN4GateLike_47081431499279
// MI455X (gfx1250) — compile-verified
//
#include <hip/hip_runtime.h>
#include <hip/hip_bf16.h>

// ---------------------------------------------------------------------------
// GNN4GateLike on MI455X (gfx1250, wave32).
// Dense linear algebra via V_WMMA_F32_16X16X4_F32 (full-fp32 matrix pipe);
// each wave computes a 16x128 output strip (8 accumulator tiles) so the A
// fragment is loaded once per k-slice and reused by 8 WMMAs -> ~4 FLOP/B
// becomes ~26 FLOP/B of L2 operand traffic. Edge scatter works on a 25.6 MB
// feature matrix resident in the 192 MB L2.
// ---------------------------------------------------------------------------

typedef __attribute__((ext_vector_type(2))) float v2f;
typedef __attribute__((ext_vector_type(8))) float v8f;

static constexpr int Nn   = 50000;   // nodes
static constexpr int Ee   = 800000;  // edges
static constexpr int Gg   = 64;      // graphs
static constexpr int HIDc = 128;
static constexpr int ENC  = 56;      // 3*16 emb + 8 numeric

// ---------------------------------------------------------------- utilities
__global__ void fill_kernel(float* __restrict__ p, float v, int n) {
  int i = blockIdx.x * blockDim.x + threadIdx.x;
  if (i < n) p[i] = v;
}

// --------------------------------------------------------------- encoder
__global__ void encode_kernel(const float* __restrict__ x,
                              const float* __restrict__ embF,
                              const float* __restrict__ embC,
                              const float* __restrict__ embK,
                              float* __restrict__ h0, int n) {
  int i = blockIdx.x * blockDim.x + threadIdx.x;
  if (i >= n) return;
  const float* xi = x + (size_t)i * 11;
  int c0 = (int)xi[0]; c0 = c0 < 0 ? 0 : (c0 > 63   ? 63   : c0);
  int c1 = (int)xi[1]; c1 = c1 < 0 ? 0 : (c1 > 1023 ? 1023 : c1);
  int c2 = (int)xi[2]; c2 = c2 < 0 ? 0 : (c2 > 31   ? 31   : c2);
  float* o = h0 + (size_t)i * ENC;
#pragma unroll
  for (int j = 0; j < 16; ++j) o[j]      = embF[c0 * 16 + j];
#pragma unroll
  for (int j = 0; j < 16; ++j) o[16 + j] = embC[c1 * 16 + j];
#pragma unroll
  for (int j = 0; j < 16; ++j) o[32 + j] = embK[c2 * 16 + j];
#pragma unroll
  for (int j = 0; j < 8;  ++j) o[48 + j] = log1pf(fmaxf(xi[3 + j], 0.0f));
}

// --------------------------------------------------------------- degrees
__global__ void degree_kernel(const int* __restrict__ src,
                              const int* __restrict__ dst,
                              float* __restrict__ degF,  // in-degree at dst
                              float* __restrict__ degB,  // out-degree at src
                              int ne) {
  int e = blockIdx.x * blockDim.x + threadIdx.x;
  if (e >= ne) return;
  atomicAdd(&degF[dst[e]], 1.0f);
  atomicAdd(&degB[src[e]], 1.0f);
}

__global__ void invert_kernel(float* __restrict__ d, int n) {
  int i = blockIdx.x * blockDim.x + threadIdx.x;
  if (i >= n) return;
  d[i] = 1.0f / fmaxf(d[i], 1.0f);
}

// -------------------------------------------------- edge scatter-add (sum)
// One thread per (edge, 4-float chunk); consecutive lanes take consecutive
// 16B chunks of the same row -> coalesced loads + L2-resident atomics.
__global__ void scatter_add_kernel(const float* __restrict__ h,
                                   const int* __restrict__ from_idx,
                                   const int* __restrict__ to_idx,
                                   float* __restrict__ acc, int ne, int F) {
  int t = blockIdx.x * blockDim.x + threadIdx.x;
  int chunks = F >> 2;
  int e = t / chunks;
  if (e >= ne) return;
  int c = (t - e * chunks) << 2;
  int s = from_idx[e], d = to_idx[e];
  const float4 v = *(const float4*)(h + (size_t)s * F + c);
  float* a = acc + (size_t)d * F + c;
  atomicAdd(a + 0, v.x); atomicAdd(a + 1, v.y);
  atomicAdd(a + 2, v.z); atomicAdd(a + 3, v.w);
}

// ---------------------------------------------------------- WMMA GEMM
// C[M x 128] = act( rs0(A0) @ W0 [+ A1 @ W1] + b0 [+ b1] )
//   A*: [M x K*] row-major (ld = lda*), W*: [K* x 128] row-major.
//   rs0: optional per-row scale applied to A0 (mean = sum * 1/deg).
// One wave computes a full 16x128 strip: 8 x (16x16) accumulator tiles.
// The A fragment (incl. 1/deg scale) is loaded once per k-slice and feeds
// 8 back-to-back WMMAs on independent accumulators (no D->A/B hazards).
// VGPR layouts per CDNA5 ISA 7.12.2:
//   A 16x4 : lane l -> row (l&15), K = k + 2*(l>>4) + {0,1}
//   B 4x16 : lane l -> col (l&15), K = k + 2*(l>>4) + {0,1}
//   C 16x16: vgpr r, lane l -> row r + 8*(l>>4), col (l&15)
__global__ __launch_bounds__(128)
void wmma_gemm_kernel(const float* __restrict__ A0, int lda0, int K0,
                      const float* __restrict__ W0,
                      const float* __restrict__ rs0,
                      const float* __restrict__ A1, int lda1, int K1,
                      const float* __restrict__ W1,
                      const float* __restrict__ b0,
                      const float* __restrict__ b1,
                      float* __restrict__ C, int ldc, int coff,
                      int M, int relu) {
  int wave = blockIdx.x * (blockDim.x >> 5) + (threadIdx.x >> 5);
  if (wave * 16 >= M) return;  // wave-uniform: EXEC stays all-1s for WMMA
  int lane = threadIdx.x & 31;
  int l15 = lane & 15, l16 = lane >> 4;
  int row = wave * 16 + l15;   // A row for this lane

  v8f acc[8];
#pragma unroll
  for (int t = 0; t < 8; ++t)
    acc[t] = (v8f){0.f, 0.f, 0.f, 0.f, 0.f, 0.f, 0.f, 0.f};

  float s0 = rs0 ? rs0[row] : 1.0f;
  const float* a0p = A0 + (size_t)row * lda0;
  for (int k = 0; k < K0; k += 4) {
    int ka = k + 2 * l16;
    v2f a;
    a.x = a0p[ka]     * s0;
    a.y = a0p[ka + 1] * s0;
    const float* w0 = W0 + (size_t)ka * HIDc + l15;        // row ka
    const float* w1 = w0 + HIDc;                           // row ka+1
#pragma unroll
    for (int t = 0; t < 8; ++t) {
      v2f b;
      b.x = w0[t * 16];
      b.y = w1[t * 16];
      acc[t] = __builtin_amdgcn_wmma_f32_16x16x4_f32(false, a, false, b,
                                                     (short)0, acc[t],
                                                     false, false);
    }
  }
  if (A1) {
    const float* a1p = A1 + (size_t)row * lda1;
    for (int k = 0; k < K1; k += 4) {
      int ka = k + 2 * l16;
      v2f a;
      a.x = a1p[ka];
      a.y = a1p[ka + 1];
      const float* w0 = W1 + (size_t)ka * HIDc + l15;
      const float* w1 = w0 + HIDc;
#pragma unroll
      for (int t = 0; t < 8; ++t) {
        v2f b;
        b.x = w0[t * 16];
        b.y = w1[t * 16];
        acc[t] = __builtin_amdgcn_wmma_f32_16x16x4_f32(false, a, false, b,
                                                       (short)0, acc[t],
                                                       false, false);
      }
    }
  }

#pragma unroll
  for (int t = 0; t < 8; ++t) {
    int col = t * 16 + l15;
    float bsum = (b0 ? b0[col] : 0.0f) + (b1 ? b1[col] : 0.0f);
#pragma unroll
    for (int r = 0; r < 8; ++r) {
      int orow = wave * 16 + r + 8 * l16;
      float v = acc[t][r] + bsum;
      if (relu) v = fmaxf(v, 0.0f);
      C[(size_t)orow * ldc + coff + col] = v;
    }
  }
}

// ----------------------------------------------------------- pooling
// sum via float atomics; max via int-bitpattern atomicMax (valid: h >= 0
// post-ReLU, and empty groups keep init 0 which matches the reference).
__global__ void pool_kernel(const float* __restrict__ h,
                            const int* __restrict__ batch,
                            float* __restrict__ psum, int* __restrict__ pmax,
                            float* __restrict__ pcnt, int n) {
  int t = blockIdx.x * blockDim.x + threadIdx.x;
  int i = t >> 5;            // 32 chunks * 4 floats = 128 features
  if (i >= n) return;
  int c = (t & 31) << 2;
  int g = batch[i];
  const float4 v = *(const float4*)(h + (size_t)i * HIDc + c);
  float* s = psum + (size_t)g * HIDc + c;
  atomicAdd(s + 0, v.x); atomicAdd(s + 1, v.y);
  atomicAdd(s + 2, v.z); atomicAdd(s + 3, v.w);
  int* m = pmax + (size_t)g * HIDc + c;
  atomicMax(m + 0, __float_as_int(v.x));
  atomicMax(m + 1, __float_as_int(v.y));
  atomicMax(m + 2, __float_as_int(v.z));
  atomicMax(m + 3, __float_as_int(v.w));
  if ((t & 31) == 0) atomicAdd(&pcnt[g], 1.0f);
}

__global__ void gfeat_kernel(const float* __restrict__ psum,
                             const int* __restrict__ pmax,
                             const float* __restrict__ pcnt,
                             float* __restrict__ gfeat) {
  int t = blockIdx.x * blockDim.x + threadIdx.x;
  if (t >= Gg * HIDc) return;
  int g = t / HIDc, f = t - g * HIDc;
  float s   = psum[(size_t)g * HIDc + f];
  float inv = 1.0f / fmaxf(pcnt[g], 1.0f);
  float* o = gfeat + (size_t)g * (3 * HIDc);
  o[f]            = s * inv;                                    // mean
  o[HIDc + f]     = __int_as_float(pmax[(size_t)g * HIDc + f]); // max
  o[2 * HIDc + f] = s;                                          // sum
}

__global__ void final_kernel(const float* __restrict__ gh,
                             const float* __restrict__ W2,
                             const float* __restrict__ b2,
                             float* __restrict__ out) {
  int g = blockIdx.x * blockDim.x + threadIdx.x;
  if (g >= Gg) return;
  float acc = b2[0];
  const float* r = gh + (size_t)g * HIDc;
#pragma unroll 8
  for (int j = 0; j < HIDc; ++j) acc += r[j] * W2[j];
  out[g] = acc;
}

// ---------------------------------------------------------------------------
extern "C" void kernel_launch(void* const* d_in, const int* in_sizes, int n_in,
                              void* d_out, int out_size, void* d_ws, size_t ws_size,
                              hipStream_t stream) {
  // setup_inputs() insertion order, recursively flattened:
  // 0 x[N,11] f32 | 1 edge_index[2,E] i32 | 2 batch[N] i32
  // 3 emb_family[64,16] | 4 emb_cell[1024,16] | 5 emb_kind[32,16]
  // 6 + 10*l + {0 fwd_Wl, 1 fwd_bl, 2 fwd_Wr, 3 fwd_br,
  //             4 bwd_Wl, 5 bwd_bl, 6 bwd_Wr, 7 bwd_br, 8 fuse_W, 9 fuse_b}
  // 36 cls_W1[384,128] | 37 cls_b1[128] | 38 cls_W2[128,1] | 39 cls_b2[1]
  const float* x     = (const float*)d_in[0];
  const int*   ei    = (const int*)d_in[1];
  const int*   batch = (const int*)d_in[2];
  const float* embF  = (const float*)d_in[3];
  const float* embC  = (const float*)d_in[4];
  const float* embK  = (const float*)d_in[5];
  const float* clsW1 = (const float*)d_in[36];
  const float* clsb1 = (const float*)d_in[37];
  const float* clsW2 = (const float*)d_in[38];
  const float* clsb2 = (const float*)d_in[39];
  const int* src = ei;        // edge_index[0]
  const int* dst = ei + Ee;   // edge_index[1]

  // workspace carve-up (all offsets 16B-aligned)
  float* ws = (float*)d_ws;
  size_t off = 0;
  auto take = [&](size_t n) { float* p = ws + off; off += n; return p; };
  float* h0    = take((size_t)Nn * ENC);
  float* hA    = take((size_t)Nn * HIDc);
  float* hB    = take((size_t)Nn * HIDc);
  float* acc   = take((size_t)Nn * HIDc);
  float* hcat  = take((size_t)Nn * 2 * HIDc);
  float* invF  = take(Nn);
  float* invB  = take(Nn);
  float* psum  = take((size_t)Gg * HIDc);
  float* pmaxf = take((size_t)Gg * HIDc);
  float* pcnt  = take(Gg);
  float* gfeat = take((size_t)Gg * 3 * HIDc);
  float* gh    = take((size_t)Gg * HIDc);
  (void)ws_size; (void)in_sizes; (void)n_in; (void)out_size;

  auto cdiv = [](long a, long b) { return (int)((a + b - 1) / b); };
  auto fill = [&](float* p, float v, long n) {
    fill_kernel<<<cdiv(n, 256), 256, 0, stream>>>(p, v, (int)n);
  };
  auto gemm = [&](const float* A0, int lda0, int K0, const float* W0,
                  const float* rs0, const float* A1, int lda1, int K1,
                  const float* W1, const float* b0, const float* b1,
                  float* C, int ldc, int coff, int M, int relu) {
    int waves = M / 16;  // one wave per 16x128 output strip
    wmma_gemm_kernel<<<cdiv(waves, 4), 128, 0, stream>>>(
        A0, lda0, K0, W0, rs0, A1, lda1, K1, W1, b0, b1, C, ldc, coff, M, relu);
  };

  // 1) encode
  encode_kernel<<<cdiv(Nn, 256), 256, 0, stream>>>(x, embF, embC, embK, h0, Nn);

  // 2) degrees -> inverse counts (clamped at 1)
  fill(invF, 0.0f, Nn);
  fill(invB, 0.0f, Nn);
  degree_kernel<<<cdiv(Ee, 256), 256, 0, stream>>>(src, dst, invF, invB, Ee);
  invert_kernel<<<cdiv(Nn, 256), 256, 0, stream>>>(invF, Nn);
  invert_kernel<<<cdiv(Nn, 256), 256, 0, stream>>>(invB, Nn);

  // 3) GNN layers
  const float* hin = h0;
  int K = ENC;
  float* hout = hA;
  for (int l = 0; l < 3; ++l) {
    const int b = 6 + 10 * l;
    const float* fwd_Wl = (const float*)d_in[b + 0];
    const float* fwd_bl = (const float*)d_in[b + 1];
    const float* fwd_Wr = (const float*)d_in[b + 2];
    const float* fwd_br = (const float*)d_in[b + 3];
    const float* bwd_Wl = (const float*)d_in[b + 4];
    const float* bwd_bl = (const float*)d_in[b + 5];
    const float* bwd_Wr = (const float*)d_in[b + 6];
    const float* bwd_br = (const float*)d_in[b + 7];
    const float* fuse_W = (const float*)d_in[b + 8];
    const float* fuse_b = (const float*)d_in[b + 9];

    long sc_threads = (long)Ee * (K >> 2);

    // forward conv: mean_{src->dst} -> hcat[:, 0:128]
    fill(acc, 0.0f, (long)Nn * K);
    scatter_add_kernel<<<cdiv(sc_threads, 256), 256, 0, stream>>>(
        hin, src, dst, acc, Ee, K);
    gemm(acc, K, K, fwd_Wl, invF, hin, K, K, fwd_Wr, fwd_bl, fwd_br,
         hcat, 2 * HIDc, 0, Nn, 0);

    // backward conv: mean_{dst->src} -> hcat[:, 128:256]
    fill(acc, 0.0f, (long)Nn * K);
    scatter_add_kernel<<<cdiv(sc_threads, 256), 256, 0, stream>>>(
        hin, dst, src, acc, Ee, K);
    gemm(acc, K, K, bwd_Wl, invB, hin, K, K, bwd_Wr, bwd_bl, bwd_br,
         hcat, 2 * HIDc, HIDc, Nn, 0);

    // fuse: h = relu(hcat @ fuse_W + fuse_b)
    gemm(hcat, 2 * HIDc, 2 * HIDc, fuse_W, nullptr,
         nullptr, 0, 0, nullptr, fuse_b, nullptr, hout, HIDc, 0, Nn, 1);

    hin = hout;
    K = HIDc;
    hout = (hout == hA) ? hB : hA;
  }

  // 4) pooling (mean | max | sum)
  fill(psum, 0.0f, (long)Gg * HIDc);
  fill(pmaxf, 0.0f, (long)Gg * HIDc);  // bits(0.0f) == 0
  fill(pcnt, 0.0f, Gg);
  pool_kernel<<<cdiv((long)Nn * 32, 256), 256, 0, stream>>>(
      hin, batch, psum, (int*)pmaxf, pcnt, Nn);
  gfeat_kernel<<<cdiv(Gg * HIDc, 256), 256, 0, stream>>>(
      psum, (const int*)pmaxf, pcnt, gfeat);

  // 5) classifier head
  gemm(gfeat, 3 * HIDc, 3 * HIDc, clsW1, nullptr,
       nullptr, 0, 0, nullptr, clsb1, nullptr, gh, HIDc, 0, Gg, 1);
  final_kernel<<<cdiv(Gg, 64), 64, 0, stream>>>(gh, clsW2, clsb2, (float*)d_out);
}